// MinCutRefinement_75161927680692
// MI455X (gfx1250) — compile-verified
//
#include <hip/hip_runtime.h>
#include <hip/hip_bf16.h>
#include <cstdint>

typedef __attribute__((ext_vector_type(2))) float v2f;
typedef __attribute__((ext_vector_type(8))) float v8f;

#define D_FEAT 128
#define K_SEG  16

// ---------------- zero scratch ----------------
__global__ void mc_zero_kernel(float* __restrict__ p, int n) {
  for (int i = blockIdx.x * blockDim.x + threadIdx.x; i < n;
       i += gridDim.x * blockDim.x)
    p[i] = 0.0f;
}

// ---------------- S = softmax(x @ W + b), WMMA f32 16x16x4 ----------------
// One wave handles a 16-row tile: M=16, N=16 (segments), K=128 in 4-wide steps.
__global__ void mc_segpred_kernel(const float* __restrict__ x,
                                  const float* __restrict__ W,
                                  const float* __restrict__ b,
                                  float* __restrict__ Sws,
                                  float* __restrict__ Sout,
                                  int N) {
  const int lane   = threadIdx.x & 31;
  const int waveId = threadIdx.x >> 5;
  const int tile   = blockIdx.x * (blockDim.x >> 5) + waveId;
  const int rowBase = tile * 16;
  if (rowBase >= N) return;

  const int n    = lane & 15;   // segment index (N dim) / A-row within tile
  const int half = lane >> 4;   // lane half selects K offset (A/B) and M offset (C/D)

  int rowA = rowBase + n;       // A-operand: lane holds row M = lane&15
  if (rowA >= N) rowA = N - 1;  // clamp for tail safety (N%16==0 in practice)
  const float* __restrict__ xrow = x + (size_t)rowA * D_FEAT;

  v8f c = {};
  #pragma unroll
  for (int k0 = 0; k0 < D_FEAT; k0 += 4) {
    const int kk = k0 + (half << 1);           // lanes 0-15: K=k0,k0+1 ; 16-31: K=k0+2,k0+3
    v2f a = *(const v2f*)(xrow + kk);          // A: x[rowA][kk], x[rowA][kk+1]
    v2f bb;
    bb.x = W[kk * K_SEG + n];                  // B: W[kk][n]
    bb.y = W[(kk + 1) * K_SEG + n];            //    W[kk+1][n]
    c = __builtin_amdgcn_wmma_f32_16x16x4_f32(false, a, false, bb,
                                              (short)0, c, false, false);
  }

  const float bias = b[n];
  #pragma unroll
  for (int j = 0; j < 8; ++j) c[j] += bias;

  // Row-wise softmax over the 16 segments: each C row lives across 16 lanes
  // of one half-wave (xor offsets 1..8 stay within the half).
  #pragma unroll
  for (int j = 0; j < 8; ++j) {
    float v = c[j];
    float m = v;
    #pragma unroll
    for (int off = 8; off >= 1; off >>= 1)
      m = fmaxf(m, __shfl_xor(m, off, 32));
    float e = __expf(v - m);
    float s = e;
    #pragma unroll
    for (int off = 8; off >= 1; off >>= 1)
      s += __shfl_xor(s, off, 32);
    float sm = e / s;
    const int row = rowBase + j + half * 8;    // D: VGPR j -> M=j (lanes 0-15) / j+8
    if (row < N) {
      Sws [(size_t)row * K_SEG + n] = sm;
      Sout[(size_t)row * K_SEG + n] = sm;
    }
  }
}

// ---------------- per-edge: w, degree scatter, cut accumulation ----------------
// One wave per edge (grid-stride). 32 lanes x float4 = coalesced 512B row loads.
__global__ void mc_edge_kernel(const float* __restrict__ x,
                               const int* __restrict__ ei,
                               const float* __restrict__ S,
                               float* __restrict__ degree,
                               float* __restrict__ cutK,
                               int E) {
  __shared__ float red[8 * K_SEG];
  const int lane  = threadIdx.x & 31;
  const int wId   = threadIdx.x >> 5;
  const int gwave = (blockIdx.x * blockDim.x + threadIdx.x) >> 5;
  const int nwave = (gridDim.x * blockDim.x) >> 5;

  float acc = 0.0f;  // lanes 0..15 accumulate cut[k=lane]
  for (int e = gwave; e < E; e += nwave) {
    const int src = ei[e];
    const int tgt = ei[E + e];
    const float4* __restrict__ xs = (const float4*)(x + (size_t)src * D_FEAT);
    const float4* __restrict__ xt = (const float4*)(x + (size_t)tgt * D_FEAT);
    float4 a = xs[lane];
    float4 t = xt[lane];
    float dx = a.x - t.x, dy = a.y - t.y, dz = a.z - t.z, dw = a.w - t.w;
    float d = dx * dx + dy * dy + dz * dz + dw * dw;
    #pragma unroll
    for (int off = 16; off >= 1; off >>= 1)
      d += __shfl_xor(d, off, 32);           // all lanes end with full dist^2
    const float w = __expf(-0.5f * d);
    if (lane == 0) atomicAdd(&degree[src], w);
    if (lane < K_SEG) {
      float Ss = S[(size_t)src * K_SEG + lane];
      float St = S[(size_t)tgt * K_SEG + lane];
      acc += w * Ss * (1.0f - St);
    }
  }
  if (lane < K_SEG) red[wId * K_SEG + lane] = acc;
  __syncthreads();
  if (threadIdx.x < K_SEG) {
    float s = 0.0f;
    const int nw = blockDim.x >> 5;
    for (int wv = 0; wv < nw; ++wv) s += red[wv * K_SEG + threadIdx.x];
    atomicAdd(&cutK[threadIdx.x], s);
  }
}

// ---------------- assoc[k] = sum_i degree[i] * S[i,k] ----------------
// Each wave processes 2 nodes/iter: lanes 0-15 -> node n0, lanes 16-31 -> n0+1.
__global__ void mc_assoc_kernel(const float* __restrict__ S,
                                const float* __restrict__ degree,
                                float* __restrict__ assocK,
                                int N) {
  __shared__ float red[8 * K_SEG];
  const int lane  = threadIdx.x & 31;
  const int wId   = threadIdx.x >> 5;
  const int gwave = (blockIdx.x * blockDim.x + threadIdx.x) >> 5;
  const int nwave = (gridDim.x * blockDim.x) >> 5;
  const int half  = lane >> 4;
  const int k     = lane & 15;

  float acc = 0.0f;
  for (int n0 = gwave * 2; n0 < N; n0 += nwave * 2) {
    const int node = n0 + half;
    if (node < N)
      acc += degree[node] * S[(size_t)node * K_SEG + k];
  }
  acc += __shfl_xor(acc, 16, 32);            // fold halves: both held k=lane&15
  if (lane < K_SEG) red[wId * K_SEG + lane] = acc;
  __syncthreads();
  if (threadIdx.x < K_SEG) {
    float s = 0.0f;
    const int nw = blockDim.x >> 5;
    for (int wv = 0; wv < nw; ++wv) s += red[wv * K_SEG + threadIdx.x];
    atomicAdd(&assocK[threadIdx.x], s);
  }
}

// ---------------- loss = sum_k (assoc>eps ? cut/assoc : 0) ----------------
__global__ void mc_final_kernel(const float* __restrict__ cutK,
                                const float* __restrict__ assocK,
                                float* __restrict__ out) {
  const int lane = threadIdx.x;
  float v = 0.0f;
  if (lane < K_SEG) {
    const float a = assocK[lane];
    v = (a > 1e-8f) ? (cutK[lane] / a) : 0.0f;
  }
  #pragma unroll
  for (int off = 16; off >= 1; off >>= 1)
    v += __shfl_xor(v, off, 32);
  if (lane == 0) out[0] = v;
}

extern "C" void kernel_launch(void* const* d_in, const int* in_sizes, int n_in,
                              void* d_out, int out_size, void* d_ws, size_t ws_size,
                              hipStream_t stream) {
  const float* x  = (const float*)d_in[0];   // (N, 128)
  const int*   ei = (const int*)d_in[1];     // (2, E) flattened: src row then tgt row
  // d_in[2] = num_expected_segments scalar (K_SEG=16, compile-time)
  const float* W  = (const float*)d_in[3];   // (128, 16)
  const float* b  = (const float*)d_in[4];   // (16,)

  const int N = in_sizes[0] / D_FEAT;
  const int E = in_sizes[1] / 2;

  float* out = (float*)d_out;                // [loss, S(N*16)...]
  float* Sws    = (float*)d_ws;              // aligned copy of S for edge pass
  float* degree = Sws + (size_t)N * K_SEG;   // N floats
  float* cutK   = degree + N;                // 16 floats
  float* assocK = cutK + K_SEG;              // 16 floats

  // degree, cutK, assocK are contiguous: zero them in one pass
  mc_zero_kernel<<<256, 256, 0, stream>>>(degree, N + 2 * K_SEG);

  const int tiles  = (N + 15) / 16;
  const int blocks = (tiles + 7) / 8;        // 8 waves (tiles) per 256-thread block
  mc_segpred_kernel<<<blocks, 256, 0, stream>>>(x, W, b, Sws, out + 1, N);

  mc_edge_kernel<<<4096, 256, 0, stream>>>(x, ei, Sws, degree, cutK, E);
  mc_assoc_kernel<<<1024, 256, 0, stream>>>(Sws, degree, assocK, N);
  mc_final_kernel<<<1, 32, 0, stream>>>(cutK, assocK, out);
}